// YOLOv2Loss_27917287424000
// MI455X (gfx1250) — compile-verified
//
#include <hip/hip_runtime.h>

// YOLOv2 loss on gfx1250 (MI455X). Memory-bound: ~104MB read -> ~4.5us floor
// at 23.3 TB/s. Reduction uses V_WMMA_F32_16X16X4_F32 (sum-by-ones matmul).

typedef __attribute__((ext_vector_type(2))) float v2f;
typedef __attribute__((ext_vector_type(8))) float v8f;

#define GRID_S   13
#define ANCH     5
#define NCLS     20
#define CELL_LAB 25
#define CELL_PRD 125

// Wave32 full reduction using one V_WMMA_F32_16X16X4_F32.
// A-matrix layout (16x4 f32, 2 VGPRs): lanes 0-15 -> {K=0,K=1}, lanes 16-31 -> {K=2,K=3}.
// With a.x = v[lane], a.y = 0 and B = all-ones (layout-independent):
//   D[m][n] = v[m] + v[m+16]  for all n.
// C/D layout: lane l (l<16) VGPR g = D[g][l]; lane l (>=16) VGPR g = D[g+8][l-16].
// Per-lane sum of the 8 D VGPRs gives: lanes 0-15 -> sum_{m=0..7}(v[m]+v[m+16]),
// lanes 16-31 -> sum_{m=8..15}(v[m]+v[m+16]). One xor-16 shuffle completes the sum.
// PRECONDITION: EXEC all ones (call from fully-converged code only).
__device__ __forceinline__ float wave_reduce_wmma(float v) {
    v2f a; a[0] = v;     a[1] = 0.0f;
    v2f b; b[0] = 1.0f;  b[1] = 1.0f;
    v8f c = {};
    v8f d = __builtin_amdgcn_wmma_f32_16x16x4_f32(
        /*neg_a=*/false, a, /*neg_b=*/false, b,
        /*c_mod=*/(short)0, c, /*reuse_a=*/false, /*reuse_b=*/false);
    float s = ((d[0] + d[1]) + (d[2] + d[3])) + ((d[4] + d[5]) + (d[6] + d[7]));
    s += __shfl_xor(s, 16, 32);   // combine the two half-sums; all lanes get total
    return s;
}

__global__ __launch_bounds__(256) void yolo_loss_main(
        const float* __restrict__ preds,
        const float* __restrict__ labels,
        float* __restrict__ partials,
        int num_cells) {
    const int tid = blockIdx.x * 256 + threadIdx.x;
    // Clamp (no divergence) so a ragged tail never breaks EXEC for the WMMA.
    const int   cid   = tid < num_cells ? tid : (num_cells - 1);
    const float valid = tid < num_cells ? 1.0f : 0.0f;

    const float* lab = labels + (size_t)cid * CELL_LAB;
    const float* prd = preds  + (size_t)cid * CELL_PRD;

    // ---- label box ----
    const float lx = lab[0], ly = lab[1], lw = lab[2], lh = lab[3];
    const float objf = (lab[4] == 1.0f) ? 1.0f : 0.0f;

    const float bx1 = lx - 0.5f * lw, bx2 = lx + 0.5f * lw;
    const float by1 = ly - 0.5f * lh, by2 = ly + 0.5f * lh;
    const float barea = lw * lh;

    // ---- pass 1: IoU + conf per anchor (5 floats each) ----
    float iou[ANCH], pconf[ANCH];
#pragma unroll
    for (int a = 0; a < ANCH; ++a) {
        const float* p = prd + a * CELL_LAB;
        const float px = p[0], py = p[1], pw = p[2], ph = p[3];
        pconf[a] = p[4];
        const float ax1 = px - 0.5f * pw, ax2 = px + 0.5f * pw;
        const float ay1 = py - 0.5f * ph, ay2 = py + 0.5f * ph;
        float iw = fminf(ax2, bx2) - fmaxf(ax1, bx1); iw = fmaxf(iw, 0.0f);
        float ih = fminf(ay2, by2) - fmaxf(ay1, by1); ih = fmaxf(ih, 0.0f);
        const float inter = iw * ih;
        const float uni   = pw * ph + barea - inter;
        iou[a] = inter / (uni + 1e-12f);
    }

    // argmax, first-max wins (matches jnp.argmax)
    int best = 0;
#pragma unroll
    for (int a = 1; a < ANCH; ++a) best = (iou[a] > iou[best]) ? a : best;

    // ---- pass 2: best anchor full record ----
    const float* bp = prd + best * CELL_LAB;
    const float bxp = bp[0], byp = bp[1], bwp = bp[2], bhp = bp[3], bcf = bp[4];

    float loss = 5.0f * ((lx - bxp) * (lx - bxp) + (ly - byp) * (ly - byp));
    const float dw = sqrtf(lw) - sqrtf(bwp);
    const float dh = sqrtf(lh) - sqrtf(bhp);
    loss += dw * dw + dh * dh;

    float bi = iou[best];
    loss += (bi - bcf) * (bi - bcf);

    float cls = 0.0f;
#pragma unroll
    for (int k = 0; k < NCLS; ++k) {
        const float d = lab[5 + k] - bp[5 + k];
        cls += d * d;
    }
    loss += cls;

    // no-obj: non-best anchors with IoU < 0.6, object cells only
    float noobj = 0.0f;
#pragma unroll
    for (int a = 0; a < ANCH; ++a) {
        const float m = (a != best && iou[a] < 0.6f) ? 1.0f : 0.0f;
        const float d = iou[a] - pconf[a];
        noobj += m * d * d;
    }

    loss = valid * objf * (loss + noobj);

    // ---- fully-converged here: WMMA wave reduce, then per-block sum ----
    const float wsum = wave_reduce_wmma(loss);

    __shared__ float lds[8];
    const int lane = threadIdx.x & 31;
    const int wave = threadIdx.x >> 5;
    if (lane == 0) lds[wave] = wsum;
    __syncthreads();
    if (threadIdx.x == 0) {
        float t = 0.0f;
#pragma unroll
        for (int w = 0; w < 8; ++w) t += lds[w];
        partials[blockIdx.x] = t;
    }
}

__global__ __launch_bounds__(256) void yolo_loss_final(
        const float* __restrict__ partials,
        float* __restrict__ out,
        int n, float inv_batch) {
    // Fixed-order strided accumulation: deterministic across replays.
    float v = 0.0f;
    for (int i = threadIdx.x; i < n; i += 256) v += partials[i];

    // Reconverged here; EXEC all ones for the WMMA reduce.
    const float wsum = wave_reduce_wmma(v);

    __shared__ float lds[8];
    const int lane = threadIdx.x & 31;
    const int wave = threadIdx.x >> 5;
    if (lane == 0) lds[wave] = wsum;
    __syncthreads();
    if (threadIdx.x == 0) {
        float t = 0.0f;
#pragma unroll
        for (int w = 0; w < 8; ++w) t += lds[w];
        out[0] = t * inv_batch;
    }
}

extern "C" void kernel_launch(void* const* d_in, const int* in_sizes, int n_in,
                              void* d_out, int out_size, void* d_ws, size_t ws_size,
                              hipStream_t stream) {
    (void)n_in; (void)out_size; (void)ws_size;
    const float* preds  = (const float*)d_in[0];   // [B,13,13,5,25] f32
    const float* labels = (const float*)d_in[1];   // [B,13,13,25]   f32
    float* out      = (float*)d_out;
    float* partials = (float*)d_ws;

    const int num_cells = in_sizes[1] / CELL_LAB;          // B*S*S = 173056
    const int batch     = num_cells / (GRID_S * GRID_S);   // 1024
    const int blocks    = (num_cells + 255) / 256;         // 676 (exact fit)

    yolo_loss_main<<<blocks, 256, 0, stream>>>(preds, labels, partials, num_cells);
    yolo_loss_final<<<1, 256, 0, stream>>>(partials, out, blocks, 1.0f / (float)batch);
}